// _HomoSAGE_64828236366583
// MI455X (gfx1250) — compile-verified
//
#include <hip/hip_runtime.h>
#include <hip/hip_bf16.h>

typedef float v2f __attribute__((ext_vector_type(2)));
typedef float v8f __attribute__((ext_vector_type(8)));

#define FEAT 64

// ---------------------------------------------------------------------------
// Kernel A: edge scatter-add aggregation.
// 16 threads per edge; each thread moves 4 contiguous floats (float4 gather
// from h[src], 4x global_atomic_add_f32 into agg[dst]). Thread 0 of each edge
// optionally bumps the in-degree count (degree is identical for both layers,
// so we only count on layer 1).
// Both the gather source (25.6MB) and scatter target (25.6MB) fit in the
// 192MB L2, so the random traffic stays on-chip.
// ---------------------------------------------------------------------------
__global__ __launch_bounds__(256) void sage_aggregate(
    const float* __restrict__ h,
    const int* __restrict__ src,
    const int* __restrict__ dst,
    float* __restrict__ agg,
    float* __restrict__ cnt,
    int n_edges, int do_count) {
  const long long t = (long long)blockIdx.x * blockDim.x + threadIdx.x;
  const long long e = t >> 4;
  if (e >= n_edges) return;
  const int q = (int)(t & 15);
  const int s = src[e];
  const int d = dst[e];
  const float4 v = *(const float4*)(h + (long long)s * FEAT + q * 4);
  float* base = agg + (long long)d * FEAT + q * 4;
  atomicAdd(base + 0, v.x);
  atomicAdd(base + 1, v.y);
  atomicAdd(base + 2, v.z);
  atomicAdd(base + 3, v.w);
  if (do_count && q == 0) atomicAdd(cnt + d, 1.0f);
}

// ---------------------------------------------------------------------------
// Kernel B: fused  out = relu( (agg/max(cnt,1)) @ Wl^T + bias + h @ Wr^T )
// One wave32 computes a 16-node x 64-feature output block using fp32 WMMA
// (V_WMMA_F32_16X16X4_F32): 16 K-steps x 4 N-tiles x 2 matrices = 128 WMMAs.
//
// Fragment layouts (CDNA5 ISA 7.12.2, 32-bit operands, wave32):
//   A (16x4, M x K):  lane m = lane&15, holds K = 2*(lane>>4) + {0,1} in v0,v1
//   B (4x16, K x N):  lane n = lane&15, holds K = 2*(lane>>4) + {0,1} in v0,v1
//   C/D (16x16):      lane n = lane&15, VGPR v holds row m = v + 8*(lane>>4)
// Row-major [N,64] data therefore loads A-frags as per-lane float2, and
// row-major Wl[o][f] loads B-frags (B = Wl^T) as contiguous float2 per lane.
// ---------------------------------------------------------------------------
__global__ __launch_bounds__(256) void sage_combine(
    const float* __restrict__ h,     // [N,64] self features
    const float* __restrict__ agg,   // [N,64] summed neighbor features
    const float* __restrict__ cnt,   // [N]    in-degree (float)
    const float* __restrict__ Wl,    // [64,64] row-major [out][in]
    const float* __restrict__ bias,  // [64]
    const float* __restrict__ Wr,    // [64,64] row-major [out][in]
    float* __restrict__ out,         // [N,64]
    int n_nodes) {
  const int lane  = threadIdx.x & 31;
  const int wave  = threadIdx.x >> 5;
  const int m     = lane & 15;   // A-frag row / B,C-frag column
  const int khalf = lane >> 4;   // 0 or 1

  const long long node = ((long long)blockIdx.x * 8 + wave) * 16;
  if (node >= n_nodes) return;   // wave-uniform: EXEC stays all-ones for WMMA

  const float c  = cnt[node + m];
  const float rc = 1.0f / fmaxf(c, 1.0f);

  const float* hrow = h   + (node + m) * FEAT;
  const float* arow = agg + (node + m) * FEAT;

  v8f acc[4];
  acc[0] = (v8f){}; acc[1] = (v8f){}; acc[2] = (v8f){}; acc[3] = (v8f){};

#pragma unroll
  for (int kk = 0; kk < 16; ++kk) {
    const int k0 = kk * 4 + khalf * 2;
    const v2f a_x = *(const v2f*)(hrow + k0);
    const v2f t   = *(const v2f*)(arow + k0);
    v2f a_mean;
    a_mean.x = t.x * rc;
    a_mean.y = t.y * rc;

#pragma unroll
    for (int t4 = 0; t4 < 4; ++t4) {
      const int n = t4 * 16 + m;
      const v2f bl = *(const v2f*)(Wl + n * FEAT + k0);  // B = Wl^T
      const v2f br = *(const v2f*)(Wr + n * FEAT + k0);  // B = Wr^T
      acc[t4] = __builtin_amdgcn_wmma_f32_16x16x4_f32(
          false, a_mean, false, bl, (short)0, acc[t4], false, false);
      acc[t4] = __builtin_amdgcn_wmma_f32_16x16x4_f32(
          false, a_x,    false, br, (short)0, acc[t4], false, false);
    }
  }

  // Epilogue: + bias, ReLU, store per documented C/D layout.
#pragma unroll
  for (int t4 = 0; t4 < 4; ++t4) {
    const float bv = bias[t4 * 16 + m];
#pragma unroll
    for (int v = 0; v < 8; ++v) {
      const int row = v + 8 * khalf;
      float val = acc[t4][v] + bv;
      val = fmaxf(val, 0.0f);
      out[(node + row) * FEAT + t4 * 16 + m] = val;
    }
  }
}

// ---------------------------------------------------------------------------
// Launch: zero scratch -> layer1 aggregate -> layer1 combine (h1 in ws)
//         -> re-zero agg -> layer2 aggregate -> layer2 combine (d_out).
// ---------------------------------------------------------------------------
extern "C" void kernel_launch(void* const* d_in, const int* in_sizes, int n_in,
                              void* d_out, int out_size, void* d_ws, size_t ws_size,
                              hipStream_t stream) {
  const float* x   = (const float*)d_in[0];
  const int*   ei  = (const int*)d_in[1];
  const float* Wl1 = (const float*)d_in[2];
  const float* b1  = (const float*)d_in[3];
  const float* Wr1 = (const float*)d_in[4];
  const float* Wl2 = (const float*)d_in[5];
  const float* b2  = (const float*)d_in[6];
  const float* Wr2 = (const float*)d_in[7];
  float* out = (float*)d_out;

  const int N = in_sizes[0] / FEAT;
  const int E = in_sizes[1] / 2;
  const int* src = ei;
  const int* dst = ei + E;

  char* ws = (char*)d_ws;
  float* agg = (float*)ws;                                            // N*64 f32
  float* cnt = (float*)(ws + (size_t)N * FEAT * sizeof(float));       // N f32
  float* h1  = (float*)(ws + ((size_t)N * FEAT + N) * sizeof(float)); // N*64 f32

  const size_t agg_bytes = (size_t)N * FEAT * sizeof(float);

  hipMemsetAsync(agg, 0, agg_bytes, stream);
  hipMemsetAsync(cnt, 0, (size_t)N * sizeof(float), stream);

  const long long threadsA = (long long)E * 16;
  const unsigned gridA = (unsigned)((threadsA + 255) / 256);
  const unsigned gridB = (unsigned)((N + 127) / 128);

  // Layer 1
  sage_aggregate<<<gridA, 256, 0, stream>>>(x, src, dst, agg, cnt, E, 1);
  sage_combine<<<gridB, 256, 0, stream>>>(x, agg, cnt, Wl1, b1, Wr1, h1, N);

  // Layer 2 (degree counts unchanged; don't re-count)
  hipMemsetAsync(agg, 0, agg_bytes, stream);
  sage_aggregate<<<gridA, 256, 0, stream>>>(h1, src, dst, agg, cnt, E, 0);
  sage_combine<<<gridB, 256, 0, stream>>>(h1, agg, cnt, Wl2, b2, Wr2, out, N);
}